// GradientEdgeLoss_60473139527866
// MI455X (gfx1250) — compile-verified
//
#include <hip/hip_runtime.h>
#include <hip/hip_bf16.h>

// GradientEdgeLoss for MI455X (gfx1250, wave32).
//
// loss = mean|sobelx(grayd)| + mean|sobely(grayd)|,  grayd = gray(pred)-gray(target)
// (conv linearity lets us Sobel the difference image once; Sobel kernels are
//  point-antisymmetric so correlation-vs-convolution only flips sign -> |.| invariant).
//
// Memory-bound: 201 MB read / 23.3 TB/s ~= 8.6 us floor. Kernel streams each
// input byte once per 16-row band (12.5% halo overhead), stages the gray-diff
// tile in LDS, and does a deterministic two-stage reduction. Block sums use
// V_WMMA_F32_16X16X4_F32 with an all-ones B matrix (sum-by-matmul).

typedef __attribute__((ext_vector_type(2))) float v2f;
typedef __attribute__((ext_vector_type(8))) float v8f;

#define GEL_H   512
#define GEL_W   512
#define GEL_HW  (GEL_H * GEL_W)
#define GEL_R   16                 // output rows per block
#define GEL_GROW (GEL_R + 2)       // 18 staged gray rows (halo)
#define GEL_GSTR (GEL_W + 2)       // 514: col-padded row stride
#define GEL_NBLK_ROW (GEL_H / GEL_R)  // 32 row-bands
#define GEL_B   32

__global__ __launch_bounds__(256)
void gel_main_kernel(const float* __restrict__ pred,
                     const float* __restrict__ targ,
                     float* __restrict__ partials) {
    __shared__ float gtile[GEL_GROW * GEL_GSTR];  // gray-diff, zero-padded cols
    __shared__ float part[256];

    const int tid = threadIdx.x;
    const int b   = blockIdx.y;
    const int r0  = blockIdx.x * GEL_R;

    const float* pb = pred + (size_t)b * 3 * GEL_HW;
    const float* tb = targ + (size_t)b * 3 * GEL_HW;

    // ---- Stage 1: stream RGB (b128 loads), fuse gray+diff into LDS ----
    // 18 rows * 128 float4-groups = 2304 groups; 9 per thread.
    for (int i = tid; i < GEL_GROW * (GEL_W / 4); i += 256) {
        const int row = i >> 7;            // / 128
        const int c   = (i & 127) << 2;    // * 4
        const int gr  = r0 - 1 + row;      // global image row (halo'd)
        float4 gd = make_float4(0.f, 0.f, 0.f, 0.f);
        if (gr >= 0 && gr < GEL_H) {
            const float* pr = pb + gr * GEL_W + c;
            const float* tr = tb + gr * GEL_W + c;
            const float4 p0 = *(const float4*)(pr);
            const float4 p1 = *(const float4*)(pr + GEL_HW);
            const float4 p2 = *(const float4*)(pr + 2 * GEL_HW);
            const float4 t0 = *(const float4*)(tr);
            const float4 t1 = *(const float4*)(tr + GEL_HW);
            const float4 t2 = *(const float4*)(tr + 2 * GEL_HW);
            gd.x = 0.299f*(p0.x-t0.x) + 0.587f*(p1.x-t1.x) + 0.114f*(p2.x-t2.x);
            gd.y = 0.299f*(p0.y-t0.y) + 0.587f*(p1.y-t1.y) + 0.114f*(p2.y-t2.y);
            gd.z = 0.299f*(p0.z-t0.z) + 0.587f*(p1.z-t1.z) + 0.114f*(p2.z-t2.z);
            gd.w = 0.299f*(p0.w-t0.w) + 0.587f*(p1.w-t1.w) + 0.114f*(p2.w-t2.w);
        }
        float* grow = &gtile[row * GEL_GSTR + 1 + c];
        grow[0] = gd.x; grow[1] = gd.y; grow[2] = gd.z; grow[3] = gd.w;
    }
    // zero-pad left/right halo columns (same-padding of the conv)
    if (tid < GEL_GROW * 2) {
        const int row = tid >> 1;
        gtile[row * GEL_GSTR + ((tid & 1) ? (GEL_W + 1) : 0)] = 0.f;
    }
    __syncthreads();

    // ---- Stage 2: 3x3 Sobel stencil from LDS, accumulate |gx|+|gy| ----
    float acc = 0.f;
    for (int i = tid; i < GEL_R * GEL_W; i += 256) {
        const int orow = i >> 9;           // / 512
        const int c    = i & (GEL_W - 1);
        const float* g0 = &gtile[orow * GEL_GSTR + c];   // (r-1, c-1)
        const float* g1 = g0 + GEL_GSTR;
        const float* g2 = g1 + GEL_GSTR;
        const float a00 = g0[0], a01 = g0[1], a02 = g0[2];
        const float a10 = g1[0],             a12 = g1[2];
        const float a20 = g2[0], a21 = g2[1], a22 = g2[2];
        const float gx = (a02 - a00) + 2.f * (a12 - a10) + (a22 - a20);
        const float gy = (a20 - a00) + 2.f * (a21 - a01) + (a22 - a02);
        acc += fabsf(gx) + fabsf(gy);
    }

    // ---- Stage 3: block reduction of 256 partials via WMMA x ones ----
    part[tid] = acc;
    __syncthreads();
    if (tid < 32) {                         // wave 0, EXEC all-ones
        v8f d = {};
        v2f bones; bones.x = 1.f; bones.y = 1.f;   // all-ones 4x16 B matrix
        const int i4 = (tid & 15) * 4 + (tid >> 4) * 2;  // A: 16x4 f32 layout
        #pragma unroll
        for (int q = 0; q < 4; ++q) {       // 4 chunks x 64 values = 256
            v2f a;
            a.x = part[q * 64 + i4];
            a.y = part[q * 64 + i4 + 1];
            // D = A x Ones + D : every D row = row-sum of A (layout-agnostic total)
            d = __builtin_amdgcn_wmma_f32_16x16x4_f32(
                    false, a, false, bones, (short)0, d, false, false);
        }
        // lanes 0..15 hold D rows 0..7 (col=lane), lanes 16..31 rows 8..15
        float s = d[0] + d[1] + d[2] + d[3] + d[4] + d[5] + d[6] + d[7];
        s += __shfl_xor(s, 16, 32);
        if (tid == 0)
            partials[blockIdx.y * GEL_NBLK_ROW + blockIdx.x] = s;
    }
}

__global__ __launch_bounds__(256)
void gel_reduce_kernel(const float* __restrict__ partials,
                       float* __restrict__ out, int n, float scale) {
    __shared__ float wsum[8];
    const int tid = threadIdx.x;
    float s = 0.f;
    for (int i = tid; i < n; i += 256) s += partials[i];
    #pragma unroll
    for (int m = 16; m > 0; m >>= 1) s += __shfl_xor(s, m, 32);
    if ((tid & 31) == 0) wsum[tid >> 5] = s;
    __syncthreads();
    if (tid == 0) {
        float t = 0.f;
        #pragma unroll
        for (int k = 0; k < 8; ++k) t += wsum[k];
        out[0] = t * scale;
    }
}

extern "C" void kernel_launch(void* const* d_in, const int* in_sizes, int n_in,
                              void* d_out, int out_size, void* d_ws, size_t ws_size,
                              hipStream_t stream) {
    const float* pred = (const float*)d_in[0];
    const float* targ = (const float*)d_in[1];
    float* out      = (float*)d_out;
    float* partials = (float*)d_ws;         // 1024 floats of scratch

    const int nblocks = GEL_B * GEL_NBLK_ROW;   // 1024
    dim3 grid(GEL_NBLK_ROW, GEL_B);
    gel_main_kernel<<<grid, 256, 0, stream>>>(pred, targ, partials);

    const float scale = 1.0f / (float)((size_t)GEL_B * GEL_H * GEL_W);
    gel_reduce_kernel<<<1, 256, 0, stream>>>(partials, out, nblocks, scale);
}